// lossx5_73967926772311
// MI455X (gfx1250) — compile-verified
//
#include <hip/hip_runtime.h>

#define HALF   6
#define MAXC   63
#define NK     21
#define B_     64
#define C_     256
#define H_     64
#define W_     64
#define HW     (H_ * W_)

typedef __attribute__((ext_vector_type(2))) float v2f;
typedef __attribute__((ext_vector_type(4))) float f32x4;
typedef __attribute__((ext_vector_type(8))) float v8f;

// ---------------------------------------------------------------------------
// Kernel 1: channel mean for both feature tensors.  HBM-bound: 512 MB streamed
// once with non-temporal loads (keep 192MB L2 clean; fm results DO stay in L2
// for kernel 2).  One thread produces 4 consecutive hw outputs -> 512B fully
// coalesced per wave per channel step.
// ---------------------------------------------------------------------------
__global__ __launch_bounds__(256) void cdna5_chan_mean(const float* __restrict__ f1,
                                                       const float* __restrict__ f2,
                                                       float* __restrict__ fm /*[2][B][HW]*/) {
    const int tid  = blockIdx.x * blockDim.x + threadIdx.x;  // 0 .. 131071
    const int feat = tid >> 16;                              // 65536 threads/feature
    const int idx  = tid & 0xFFFF;
    const int b    = idx >> 10;                              // 1024 float4 per image
    const int hw4  = (idx & 1023) << 2;
    const float* f = feat ? f2 : f1;
    const f32x4* src = (const f32x4*)(f + (size_t)b * C_ * HW + hw4);

    f32x4 acc = {0.f, 0.f, 0.f, 0.f};
#pragma unroll 8
    for (int c = 0; c < C_; ++c) {
        f32x4 v = __builtin_nontemporal_load(src + c * (HW / 4));
        acc += v;
    }
    acc *= (1.0f / C_);
    *(f32x4*)(fm + ((size_t)feat * B_ + b) * HW + hw4) = acc;
}

// ---------------------------------------------------------------------------
// Kernel 2: per-(feature,batch) box sums via WMMA.
//   tmp[k,w] = sum_h rmask[k,h] * fm[h,w]        (V_WMMA_F32_16X16X4_F32)
//   sum[k]   = sum_w tmp[k,w]  * cmask[k,w]      (lane mask + shfl reduce)
// One wave32 per block; EXEC is all-ones at every WMMA site.
// ---------------------------------------------------------------------------
__global__ __launch_bounds__(32) void cdna5_box_wmma(const float* __restrict__ fm, /*[2][B][HW]*/
                                                     const int* __restrict__ pre1,
                                                     const int* __restrict__ pre2,
                                                     float* __restrict__ bm /*[2][B][NK]*/) {
    const int bi   = blockIdx.x;     // 0..127
    const int feat = bi >> 6;
    const int b    = bi & 63;
    const int L    = threadIdx.x;    // 0..31
    const int* pre = feat ? pre2 : pre1;
    const float* fmb = fm + ((size_t)feat * B_ + b) * HW;

    __shared__ int   s_left[32], s_right[32], s_down[32], s_up[32];
    __shared__ float s_cnt[32];
    __shared__ float s_sum[32];

    if (L < NK) {
        const int x  = pre[(b * NK + L) * 2 + 0];
        const int y  = pre[(b * NK + L) * 2 + 1];
        const int le = max(x - HALF, 0);
        const int ri = min(x + HALF, MAXC);   // exclusive
        const int dn = max(y - HALF, 0);
        const int up = min(y + HALF, MAXC);   // exclusive
        s_left[L] = le; s_right[L] = ri; s_down[L] = dn; s_up[L] = up;
        s_cnt[L]  = (float)((ri - le) * (up - dn));
    } else {
        s_left[L] = 0; s_right[L] = 0; s_down[L] = 0; s_up[L] = 0;
        s_cnt[L]  = 1.0f;
    }
    __syncthreads();

    const int half   = L >> 4;        // 0: lanes 0-15, 1: lanes 16-31
    const int lane16 = L & 15;
    const int baseK  = half * 2;      // A/B VGPR0 holds K=0 (low half) / K=2 (high half)

    // A-matrix row windows for both M-tiles (rows 0..15 and 16..31; rows >=21 are zero)
    const int l0 = s_left[lane16],      rt0 = s_right[lane16];
    const int l1 = s_left[lane16 + 16], rt1 = s_right[lane16 + 16];

    v8f acc[2][4];
    const v8f vzero = {0.f, 0.f, 0.f, 0.f, 0.f, 0.f, 0.f, 0.f};
#pragma unroll
    for (int mt = 0; mt < 2; ++mt)
#pragma unroll
        for (int nt = 0; nt < 4; ++nt)
            acc[mt][nt] = vzero;

    for (int kk = 0; kk < 16; ++kk) {           // K dimension = H, 4 at a time
        const int h0 = kk * 4 + baseK;
        const int h1 = h0 + 1;
        v2f a0, a1;
        a0.x = (h0 >= l0 && h0 < rt0) ? 1.f : 0.f;
        a0.y = (h1 >= l0 && h1 < rt0) ? 1.f : 0.f;
        a1.x = (h0 >= l1 && h0 < rt1) ? 1.f : 0.f;
        a1.y = (h1 >= l1 && h1 < rt1) ? 1.f : 0.f;
#pragma unroll
        for (int nt = 0; nt < 4; ++nt) {
            const int w = nt * 16 + lane16;
            v2f bmat;
            bmat.x = fmb[h0 * W_ + w];
            bmat.y = fmb[h1 * W_ + w];
            acc[0][nt] = __builtin_amdgcn_wmma_f32_16x16x4_f32(
                false, a0, false, bmat, (short)0, acc[0][nt], false, false);
            acc[1][nt] = __builtin_amdgcn_wmma_f32_16x16x4_f32(
                false, a1, false, bmat, (short)0, acc[1][nt], false, false);
        }
    }

    // Second contraction: sums[k] = sum_w tmp[k,w] * cmask[k,w]
    // D layout: VGPR v holds row v + 8*half (within the 16-row tile), col lane16.
#pragma unroll
    for (int mt = 0; mt < 2; ++mt) {
#pragma unroll
        for (int v = 0; v < 8; ++v) {
            const int r  = mt * 16 + v + half * 8;
            const int dn = s_down[r], up = s_up[r];
            float partial = 0.f;
#pragma unroll
            for (int nt = 0; nt < 4; ++nt) {
                const int w = nt * 16 + lane16;
                const float t = acc[mt][nt][v];
                partial += (w >= dn && w < up) ? t : 0.f;
            }
            // reduce across the 16 lanes of each half (halves stay disjoint)
#pragma unroll
            for (int off = 1; off < 16; off <<= 1)
                partial += __shfl_xor(partial, off, 32);
            if (lane16 == 0 && r < NK) s_sum[r] = partial;
        }
    }
    __syncthreads();

    if (L < NK)
        bm[((size_t)feat * B_ + b) * NK + L] = s_sum[L] / s_cnt[L];
}

// ---------------------------------------------------------------------------
// Kernel 3: batch mean + MSE scalar.
// ---------------------------------------------------------------------------
__global__ __launch_bounds__(32) void cdna5_final(const float* __restrict__ bm,
                                                  float* __restrict__ out) {
    const int k = threadIdx.x;
    float v = 0.f;
    if (k < NK) {
        float s1 = 0.f, s2 = 0.f;
        for (int b = 0; b < B_; ++b) {
            s1 += bm[(0 * B_ + b) * NK + k];
            s2 += bm[(1 * B_ + b) * NK + k];
        }
        s1 *= (1.0f / B_);
        s2 *= (1.0f / B_);
        const float d = s1 - 0.999f * s2;   // MOMENTUM * fea_c2
        v = d * d;
    }
#pragma unroll
    for (int off = 1; off < 32; off <<= 1)
        v += __shfl_xor(v, off, 32);
    if (k == 0) out[0] = v * (1.0f / NK);
}

// ---------------------------------------------------------------------------
extern "C" void kernel_launch(void* const* d_in, const int* in_sizes, int n_in,
                              void* d_out, int out_size, void* d_ws, size_t ws_size,
                              hipStream_t stream) {
    const float* f1   = (const float*)d_in[0];
    const float* f2   = (const float*)d_in[1];
    const int*   pre1 = (const int*)d_in[2];
    const int*   pre2 = (const int*)d_in[3];

    float* fm = (float*)d_ws;                     // 2 * 64 * 4096 floats = 2 MB
    float* bm = fm + (size_t)2 * B_ * HW;         // 2 * 64 * 21 floats

    // Stage 1: channel means (memory-bound, ~512 MB @ 23.3 TB/s)
    cdna5_chan_mean<<<512, 256, 0, stream>>>(f1, f2, fm);
    // Stage 2: masked einsum via V_WMMA_F32_16X16X4_F32 (L2-resident)
    cdna5_box_wmma<<<128, 32, 0, stream>>>(fm, pre1, pre2, bm);
    // Stage 3: final MSE scalar
    cdna5_final<<<1, 32, 0, stream>>>(bm, (float*)d_out);
}